// OmegaRNNMemoryCell_77189152244388
// MI455X (gfx1250) — compile-verified
//
#include <hip/hip_runtime.h>
#include <math.h>

#define TT   512
#define DIMD 512
#define NH   8
#define DHD  64
#define WINW 8
#define KCONV 4
#define EPS_RMS 1.1920929e-07f

// LDS row strides chosen for conflict-free fragment access on 64-bank LDS:
//  SST=68: A-frag column reads -> bank (4*row+col)%64, distinct per lane; 272B rows keep b64 aligned
//  AST=72: B-frag reads -> half-wave collision needs dn == (2*72)%64 == 16, impossible
#define SST 68
#define AST 72

typedef __attribute__((ext_vector_type(2))) float v2f;
typedef __attribute__((ext_vector_type(8))) float v8f;

// ---------------------------------------------------------------- RMSNorm
__global__ void k_rmsnorm(const float* __restrict__ x, const float* __restrict__ w,
                          float* __restrict__ xn) {
  int t = blockIdx.x;
  __shared__ float red[256];
  const float* xr = x + (size_t)t * DIMD;
  float s = 0.f;
  for (int i = threadIdx.x; i < DIMD; i += 256) { float v = xr[i]; s += v * v; }
  red[threadIdx.x] = s; __syncthreads();
  for (int off = 128; off > 0; off >>= 1) {
    if (threadIdx.x < off) red[threadIdx.x] += red[threadIdx.x + off];
    __syncthreads();
  }
  float scale = rsqrtf(red[0] * (1.0f / DIMD) + EPS_RMS);
  for (int i = threadIdx.x; i < DIMD; i += 256)
    xn[(size_t)t * DIMD + i] = xr[i] * scale * w[i];
}

// ------------------------------------------------- depthwise conv (K=4, pad 2, trim)
__global__ void k_dwconv3(const float* __restrict__ xn,
                          const float* __restrict__ wq, const float* __restrict__ wk,
                          const float* __restrict__ wv,
                          float* __restrict__ qin, float* __restrict__ kin,
                          float* __restrict__ vin) {
  int t = blockIdx.x;
  int c = threadIdx.x;  // blockDim = 512
  float aq = 0.f, ak = 0.f, av = 0.f;
#pragma unroll
  for (int j = 0; j < KCONV; j++) {
    int tt = t - (KCONV / 2) + j;
    if (tt >= 0 && tt < TT) {
      float xv = xn[(size_t)tt * DIMD + c];
      aq += xv * wq[c * KCONV + j];
      ak += xv * wk[c * KCONV + j];
      av += xv * wv[c * KCONV + j];
    }
  }
  qin[(size_t)t * DIMD + c] = aq;
  kin[(size_t)t * DIMD + c] = ak;
  vin[(size_t)t * DIMD + c] = av;
}

// ---------------------------------------------------------------- gates (sigmoid)
__global__ void k_gates(const float* __restrict__ xn,
                        const float* lr_w, const float* lr_b,
                        const float* de_w, const float* de_b,
                        const float* om_w, const float* om_b,
                        float* lrg, float* deg, float* omg) {
  int t = blockIdx.x;
  int g = blockIdx.y >> 3;
  int h = blockIdx.y & 7;
  const float* w = (g == 0 ? lr_w : g == 1 ? de_w : om_w) + h * DIMD;
  float bias = (g == 0 ? lr_b : g == 1 ? de_b : om_b)[h];
  float* out = (g == 0 ? lrg : g == 1 ? deg : omg);
  __shared__ float red[64];
  const float* xr = xn + (size_t)t * DIMD;
  float s = 0.f;
  for (int i = threadIdx.x; i < DIMD; i += 64) s += xr[i] * w[i];
  red[threadIdx.x] = s; __syncthreads();
  for (int off = 32; off > 0; off >>= 1) {
    if (threadIdx.x < off) red[threadIdx.x] += red[threadIdx.x + off];
    __syncthreads();
  }
  if (threadIdx.x == 0)
    out[h * TT + t] = 1.f / (1.f + __expf(-(red[0] + bias)));
}

// ------------------------------------- Y[m,n] = sum_k X[m,k]*W[n,k]; M=N=K=512, f32 WMMA
// grid 32 blocks x 256 threads; one wave per 16x64 strip (A-frag reused across 4 B-frags)
__global__ void __launch_bounds__(256) k_gemm_nt(const float* __restrict__ X,
                                                 const float* __restrict__ W,
                                                 float* __restrict__ Y) {
  int wv = (blockIdx.x << 3) + (threadIdx.x >> 5);  // 0..255
  int lane = threadIdx.x & 31;
  int mt = wv >> 3;   // 0..31  (rows mt*16 ..)
  int ns = wv & 7;    // col strip ns*64 ..
  int row  = mt * 16 + (lane & 15);
  int koff = 2 * (lane >> 4);
  v8f zero = {0.f, 0.f, 0.f, 0.f, 0.f, 0.f, 0.f, 0.f};
  v8f acc[4];
#pragma unroll
  for (int j = 0; j < 4; j++) acc[j] = zero;
  for (int kb = 0; kb < DIMD; kb += 4) {
    int k = kb + koff;
    v2f a;
    a[0] = X[(size_t)row * DIMD + k];
    a[1] = X[(size_t)row * DIMD + k + 1];
#pragma unroll
    for (int j = 0; j < 4; j++) {
      int col = ns * 64 + j * 16 + (lane & 15);
      v2f b;
      b[0] = W[(size_t)col * DIMD + k];
      b[1] = W[(size_t)col * DIMD + k + 1];
      acc[j] = __builtin_amdgcn_wmma_f32_16x16x4_f32(false, a, false, b,
                                                     (short)0, acc[j], false, false);
    }
  }
#pragma unroll
  for (int j = 0; j < 4; j++) {
    int n  = ns * 64 + j * 16 + (lane & 15);
    int mb = 8 * (lane >> 4);
#pragma unroll
    for (int r = 0; r < 8; r++)
      Y[(size_t)(mt * 16 + mb + r) * DIMD + n] = acc[j][r];
  }
}

// --------------------------------------------- per-head RMS norm + reshape to [H,T,DH]
__global__ void k_headnorm(const float* __restrict__ Yq, const float* __restrict__ Yk,
                           const float* __restrict__ Yv,
                           const float* __restrict__ qg, const float* __restrict__ kg,
                           float* __restrict__ qh, float* __restrict__ kh,
                           float* __restrict__ vh) {
  int t = blockIdx.x, h = blockIdx.y, d = threadIdx.x;  // blockDim 64
  __shared__ float rq[64], rk[64];
  size_t src = (size_t)t * DIMD + h * DHD + d;
  float qv = Yq[src], kv = Yk[src], vv = Yv[src];
  rq[d] = qv * qv; rk[d] = kv * kv; __syncthreads();
  for (int off = 32; off > 0; off >>= 1) {
    if (d < off) { rq[d] += rq[d + off]; rk[d] += rk[d + off]; }
    __syncthreads();
  }
  float qrms = fmaxf(sqrtf(rq[0]) * 0.125f, 1e-8f);
  float krms = fmaxf(sqrtf(rk[0]) * 0.125f, 1e-8f);
  size_t o = ((size_t)h * TT + t) * DHD + d;
  qh[o] = qv / qrms * qg[h * DHD + d];
  kh[o] = kv / krms * kg[h * DHD + d];
  vh[o] = vv;
}

// --------------------------------------------------------- serial affine recurrence
// One workgroup per head. Running G/C updated incrementally via WIN-deep ring buffers;
// per step S_new = S @ A + C via 16 tiles of chained V_WMMA_F32_16X16X4_F32.
// Token loads for step t+1 are issued during step t (register double-buffering) so the
// global-load latency is off the 512-step serial critical path.
__global__ void __launch_bounds__(256) k_recurrence(
    const float* __restrict__ qh, const float* __restrict__ kh,
    const float* __restrict__ vh,
    const float* __restrict__ lrg, const float* __restrict__ deg,
    const float* __restrict__ omg,
    const float* __restrict__ S0, float* __restrict__ ycat) {
  __shared__ float Sbuf[2][DHD * SST];
  __shared__ float G[DHD * DHD];
  __shared__ float Cm[DHD * SST];
  __shared__ float Amat[DHD * AST];
  __shared__ float kring[WINW][DHD], vring[WINW][DHD];
  __shared__ float sring[WINW];
  __shared__ float kcur[DHD], vcur[DHD], qcur[DHD];
  __shared__ float scal[2];

  int h = blockIdx.x;
  int tid = threadIdx.x, lane = tid & 31, wv = tid >> 5;

  for (int i = tid; i < DHD * DHD; i += 256) {
    int r = i >> 6, c = i & 63;
    Sbuf[0][r * SST + c] = S0[(size_t)h * DHD * DHD + i];
    G[i] = 0.f;
    Cm[r * SST + c] = 0.f;
  }

  // prologue: prefetch token 0 into registers
  float kreg = 0.f, vreg = 0.f, qreg = 0.f;
  float g_om = 0.f, g_lr = 0.f, g_de = 0.f;
  if (tid < DHD) {
    size_t o = ((size_t)h * TT + 0) * DHD + tid;
    kreg = kh[o]; vreg = vh[o]; qreg = qh[o];
  }
  if (tid == 0) {
    g_om = omg[h * TT]; g_lr = lrg[h * TT]; g_de = deg[h * TT];
  }
  __syncthreads();

  int iRow  = tid >> 2;          // 0..63 : owned row of G/C/A
  int jBase = (tid & 3) << 4;    // 16-wide column chunk

  for (int t = 0; t < TT; t++) {
    int cur = t & 1, nxt = cur ^ 1;

    // commit prefetched token t to LDS
    if (tid < DHD) { kcur[tid] = kreg; vcur[tid] = vreg; qcur[tid] = qreg; }
    if (tid == 0)  { scal[0] = g_om * g_lr; scal[1] = g_de; }
    __syncthreads();

    // issue loads for token t+1 (hidden behind this step's WMMA work)
    if (t + 1 < TT) {
      if (tid < DHD) {
        size_t o = ((size_t)h * TT + (t + 1)) * DHD + tid;
        kreg = kh[o]; vreg = vh[o]; qreg = qh[o];
      }
      if (tid == 0) {
        g_om = omg[h * TT + t + 1];
        g_lr = lrg[h * TT + t + 1];
        g_de = deg[h * TT + t + 1];
      }
    }

    // incremental sliding-window update of G, C; build A = decay*(I - G)
    {
      float s = scal[0], dec = scal[1];
      int r = t & (WINW - 1);
      bool sub = (t >= WINW);
      float sold  = sub ? sring[r] : 0.f;
      float ki    = kcur[iRow], vi = vcur[iRow];
      float koldi = sub ? kring[r][iRow] : 0.f;
      float voldi = sub ? vring[r][iRow] : 0.f;
#pragma unroll
      for (int jj = 0; jj < 16; jj++) {
        int j = jBase + jj;
        float kj    = kcur[j];
        float koldj = sub ? kring[r][j] : 0.f;
        float g = G[iRow * DHD + j] + s * ki * kj - sold * koldi * koldj;
        G[iRow * DHD + j] = g;
        Amat[iRow * AST + j] = dec * ((iRow == j ? 1.f : 0.f) - g);
        Cm[iRow * SST + j] = Cm[iRow * SST + j] + s * vi * kj - sold * voldi * koldj;
      }
    }
    __syncthreads();

    // stash current token into ring (not read again until t+WIN)
    if (tid < DHD) {
      int r = t & (WINW - 1);
      kring[r][tid] = kcur[tid];
      vring[r][tid] = vcur[tid];
      if (tid == 0) sring[r] = scal[0];
    }

    // S_new = S @ A + C  (16 tiles of 16x16; 2 per wave; full EXEC for WMMA)
    const float* Sread  = Sbuf[cur];
    float*       Swrite = Sbuf[nxt];
#pragma unroll
    for (int ttile = 0; ttile < 2; ttile++) {
      int tile = wv * 2 + ttile;
      int mt = tile >> 2, nt = tile & 3;
      int n    = nt * 16 + (lane & 15);
      int mrow = mt * 16 + (lane & 15);
      int mb   = 8 * (lane >> 4);
      int koff = 2 * (lane >> 4);
      v8f acc;
#pragma unroll
      for (int r2 = 0; r2 < 8; r2++) acc[r2] = Cm[(mt * 16 + mb + r2) * SST + n];
#pragma unroll
      for (int kb = 0; kb < DHD; kb += 4) {
        int k = kb + koff;
        v2f a; a[0] = Sread[mrow * SST + k]; a[1] = Sread[mrow * SST + k + 1];
        v2f b; b[0] = Amat[k * AST + n];     b[1] = Amat[(k + 1) * AST + n];
        acc = __builtin_amdgcn_wmma_f32_16x16x4_f32(false, a, false, b,
                                                    (short)0, acc, false, false);
      }
#pragma unroll
      for (int r2 = 0; r2 < 8; r2++) Swrite[(mt * 16 + mb + r2) * SST + n] = acc[r2];
    }
    __syncthreads();

    // y_t = S_t @ q_t
    if (tid < DHD) {
      float acc = 0.f;
      const float* srow = Swrite + tid * SST;
#pragma unroll 8
      for (int d = 0; d < DHD; d++) acc += srow[d] * qcur[d];
      ycat[(size_t)t * DIMD + h * DHD + tid] = acc;
    }
    __syncthreads();
  }
}

// ---------------------------------------------------------------- launch
extern "C" void kernel_launch(void* const* d_in, const int* in_sizes, int n_in,
                              void* d_out, int out_size, void* d_ws, size_t ws_size,
                              hipStream_t stream) {
  const float* x    = (const float*)d_in[0];
  const float* pnw  = (const float*)d_in[1];
  const float* Wq   = (const float*)d_in[2];
  const float* Wk   = (const float*)d_in[3];
  const float* Wv   = (const float*)d_in[4];
  const float* Wo   = (const float*)d_in[5];
  const float* qcw  = (const float*)d_in[6];
  const float* kcw  = (const float*)d_in[7];
  const float* vcw  = (const float*)d_in[8];
  const float* lr_w = (const float*)d_in[9];
  const float* lr_b = (const float*)d_in[10];
  const float* de_w = (const float*)d_in[11];
  const float* de_b = (const float*)d_in[12];
  const float* om_w = (const float*)d_in[13];
  const float* om_b = (const float*)d_in[14];
  const float* qg   = (const float*)d_in[15];
  const float* kg   = (const float*)d_in[16];
  const float* S0   = (const float*)d_in[17];
  float* out = (float*)d_out;

  const size_t TD = (size_t)TT * DIMD;  // 262144 floats
  float* ws = (float*)d_ws;
  float* buf0 = ws;            // xn, then Yq
  float* buf1 = buf0 + TD;     // qin, then qh
  float* buf2 = buf1 + TD;     // kin, then kh
  float* buf3 = buf2 + TD;     // vin, then vh
  float* buf4 = buf3 + TD;     // Yk,  then ycat
  float* buf5 = buf4 + TD;     // Yv
  float* lrg  = buf5 + TD;     // [H,T]
  float* deg  = lrg + NH * TT;
  float* omg  = deg + NH * TT;

  k_rmsnorm<<<TT, 256, 0, stream>>>(x, pnw, buf0);
  k_dwconv3<<<TT, DIMD, 0, stream>>>(buf0, qcw, kcw, vcw, buf1, buf2, buf3);
  k_gates<<<dim3(TT, 24), 64, 0, stream>>>(buf0, lr_w, lr_b, de_w, de_b,
                                           om_w, om_b, lrg, deg, omg);
  k_gemm_nt<<<32, 256, 0, stream>>>(buf2, Wk, buf4);          // Yk
  k_gemm_nt<<<32, 256, 0, stream>>>(buf3, Wv, buf5);          // Yv
  k_gemm_nt<<<32, 256, 0, stream>>>(buf1, Wq, buf0);          // Yq (xn dead now)
  k_headnorm<<<dim3(TT, NH), 64, 0, stream>>>(buf0, buf4, buf5, qg, kg,
                                              buf1, buf2, buf3);
  k_recurrence<<<NH, 256, 0, stream>>>(buf1, buf2, buf3, lrg, deg, omg, S0, buf4);
  k_gemm_nt<<<32, 256, 0, stream>>>(buf4, Wo, out);           // final projection
}